// IFNode_25280177504710
// MI455X (gfx1250) — compile-verified
//
#include <hip/hip_runtime.h>
#include <stdint.h>

// Integrate-and-Fire scan over T timesteps.
//   mem += x[t]; spike = (mem > 1.0f); mem = spike ? 0.0f : mem;
// x: [T=32, B=128, N=16384] fp32 -> spikes same shape, fp32.
//
// HBM-stream kernel (512MB one-pass, > 192MB L2 -> NT hints):
//  - each lane owns 8 consecutive neurons (2x float4, 32B) -> 2 independent
//    B128 NT loads + 2 NT stores per timestep, 1KB contiguous per wave.
//  - register double-buffer: loads for t+1 issued before compute of t, so
//    waits become partial (s_wait_loadcnt <= 2) instead of full drains.
//  - uniform SGPR base + 32-bit per-lane byte offset -> GVS addressing
//    (saddr + voffset), one v_add_nc_u32 per step instead of 64-bit adds.
//  - global_prefetch_b8 of the t+2 slab (reg buffer covers t+1).

typedef float v4f __attribute__((ext_vector_type(4)));

static constexpr int   T_STEPS = 32;
static constexpr float THRESH  = 1.0f;
static constexpr float RESETV  = 0.0f;

__device__ __forceinline__ v4f nt_load4(const char* __restrict__ base, uint32_t off) {
    return __builtin_nontemporal_load(reinterpret_cast<const v4f*>(base + off));
}
__device__ __forceinline__ void nt_store4(char* __restrict__ base, uint32_t off, v4f v) {
    __builtin_nontemporal_store(v, reinterpret_cast<v4f*>(base + off));
}

// mem += xv; spike = (mem > 1); mem = spike ? 0 : mem. Returns spike.
__device__ __forceinline__ v4f if_step(v4f& mem, v4f xv) {
    v4f spk;
    mem.x += xv.x;  mem.y += xv.y;  mem.z += xv.z;  mem.w += xv.w;
    spk.x = (mem.x > THRESH) ? 1.0f : 0.0f;
    spk.y = (mem.y > THRESH) ? 1.0f : 0.0f;
    spk.z = (mem.z > THRESH) ? 1.0f : 0.0f;
    spk.w = (mem.w > THRESH) ? 1.0f : 0.0f;
    mem.x = (mem.x > THRESH) ? RESETV : mem.x;
    mem.y = (mem.y > THRESH) ? RESETV : mem.y;
    mem.z = (mem.z > THRESH) ? RESETV : mem.z;
    mem.w = (mem.w > THRESH) ? RESETV : mem.w;
    return spk;
}

__global__ __launch_bounds__(256)
void if_scan_kernel(const float* __restrict__ xf,
                    float* __restrict__ outf,
                    unsigned n_pair,     // number of 32B lane-groups (B*N/8)
                    uint32_t stride_b)   // byte stride between timesteps (B*N*4)
{
    unsigned i = blockIdx.x * blockDim.x + threadIdx.x;
    if (i >= n_pair) return;

    const char* __restrict__ x = reinterpret_cast<const char*>(xf);
    char* __restrict__       o = reinterpret_cast<char*>(outf);

    uint32_t off = i * 32u;   // max = 256MB - 32, fits in u32

    v4f memA = {0.0f, 0.0f, 0.0f, 0.0f};
    v4f memB = {0.0f, 0.0f, 0.0f, 0.0f};

    // Prime the register double-buffer with timestep 0.
    v4f a = nt_load4(x, off);
    v4f b = nt_load4(x, off + 16u);

#pragma unroll 4
    for (int t = 0; t < T_STEPS - 1; ++t) {
        const uint32_t offn = off + stride_b;

        // L2 prefetch two steps ahead (register buffer covers t+1).
        if (t + 2 < T_STEPS)
            __builtin_prefetch((const void*)(x + (off + 2u * stride_b)), 0, 0);

        // Issue next timestep's loads before touching current data:
        // compute/store of step t overlaps the latency of step t+1.
        v4f an = nt_load4(x, offn);
        v4f bn = nt_load4(x, offn + 16u);

        v4f sa = if_step(memA, a);
        v4f sb = if_step(memB, b);
        nt_store4(o, off, sa);
        nt_store4(o, off + 16u, sb);

        a = an;  b = bn;  off = offn;
    }

    // Final (peeled) timestep.
    v4f sa = if_step(memA, a);
    v4f sb = if_step(memB, b);
    nt_store4(o, off, sa);
    nt_store4(o, off + 16u, sb);
}

extern "C" void kernel_launch(void* const* d_in, const int* in_sizes, int n_in,
                              void* d_out, int out_size, void* d_ws, size_t ws_size,
                              hipStream_t stream)
{
    const float* x   = (const float*)d_in[0];
    float*       out = (float*)d_out;

    // in_sizes[0] = T * B * N; leading dim T is the scan axis.
    long long total  = (long long)in_sizes[0];
    long long per_t  = total / T_STEPS;           // B*N = 2,097,152
    unsigned  n_pair = (unsigned)(per_t / 8);     // 262,144 lane-groups of 32B
    uint32_t  stride = (uint32_t)(per_t * 4);     // 8,388,608 bytes per timestep

    const int block = 256;                        // 8 wave32s per workgroup
    unsigned  grid  = (n_pair + block - 1) / block;

    if_scan_kernel<<<grid, block, 0, stream>>>(x, out, n_pair, stride);
}